// Correlation_77764677861501
// MI455X (gfx1250) — compile-verified
//
#include <hip/hip_runtime.h>

// Correlation volume (RAFT-style), radius 4, fp32, gfx1250.
// out[b, di*9+dj, i, j] = (1/C) * sum_c x1[b,c,i,j] * x2[b,c,i+di-4, j+dj-4]
//
// Per block (one (b, i, 16-wide j tile)): banded GEMM via V_WMMA_F32_16X16X4_F32,
// one wave per di (9 waves). x2 strips staged with GLOBAL_LOAD_ASYNC_TO_LDS_B128
// (ASYNCcnt-tracked, layout-preserving copy); zero-padding comes from a one-time
// LDS clear that async copies never overwrite. The strip is split into two
// 16-column half-arrays so the LDS load merger is forced to pair (k,k+1) B-matrix
// elements into consecutive VGPRs (no v_mov repacking before WMMA).

#define RAD   4
#define DD    9      // 2*RAD+1
#define C_DIM 256
#define H_DIM 96
#define W_DIM 128
#define B_DIM 8
#define KC    32     // K-chunk staged in LDS per iteration
#define TM    16     // output pixel tile (M dimension)
#define NW    32     // x2 strip width (two 16-wide N tiles)
#define NTHR  288    // 9 waves

typedef float v2f __attribute__((ext_vector_type(2)));
typedef float v8f __attribute__((ext_vector_type(8)));

#if __has_builtin(__builtin_amdgcn_global_load_async_to_lds_b128)
#define HAVE_ASYNC_LDS 1
// Prototype (from clang diagnostic): param 1 is <4 x i32> addrspace(1)*.
typedef int v4i_t __attribute__((vector_size(16)));
typedef __attribute__((address_space(1))) v4i_t gv4i_t;
typedef __attribute__((address_space(3))) v4i_t lv4i_t;
__device__ __forceinline__ void async_cp16(const float* g, float* l) {
  // global -> LDS, 16B per lane, tracked by ASYNCcnt. offset=0, cpol=0 (RT).
  __builtin_amdgcn_global_load_async_to_lds_b128((gv4i_t*)g, (lv4i_t*)l, 0, 0);
}
__device__ __forceinline__ void wait_async0() {
#if __has_builtin(__builtin_amdgcn_s_wait_asynccnt)
  __builtin_amdgcn_s_wait_asynccnt(0);
#else
  asm volatile("s_wait_asynccnt 0" ::: "memory");
#endif
}
#endif

__global__ __launch_bounds__(NTHR, 2)
void corr_wmma_kernel(const float* __restrict__ x1,
                      const float* __restrict__ x2,
                      float* __restrict__ out) {
  __shared__ float sA[TM][34];                      //  2.1 KB: x1 chunk [m][k]
  __shared__ __align__(16) float sB0[DD][KC][16];   // 18 KB: x2 cols 0..15  [di][k][n]
  __shared__ __align__(16) float sB1[DD][KC][16];   // 18 KB: x2 cols 16..31 [di][k][n]
  __shared__ float sD[DD][TM][NW];                  // 18 KB: result tiles

  const int jt = blockIdx.x;              // j tile: 0..7
  const int i  = blockIdx.y;              // row:    0..95
  const int b  = blockIdx.z;              // batch:  0..7
  const int j0 = jt * TM;

  const int tid  = (int)threadIdx.x;      // 0..287
  const int wave = tid >> 5;              // 0..8  == di for this wave
  const int lane = tid & 31;
  const int mm   = lane & 15;
  const int hi   = lane >> 4;             // half-wave: 0 or 1

  const float* x1b = x1 + (size_t)b * C_DIM * H_DIM * W_DIM;
  const float* x2b = x2 + (size_t)b * C_DIM * H_DIM * W_DIM;

  const int  i2       = i + wave - RAD;          // x2 row for this wave's di
  const bool rowValid = (unsigned)i2 < (unsigned)H_DIM;

  // One-time clear of the strip halves: any 16B group an async copy skips
  // (OOB row/cols) stays 0, implementing the reference's zero padding.
  {
    float4 z = make_float4(0.f, 0.f, 0.f, 0.f);
    float* base0 = &sB0[0][0][0];
    float* base1 = &sB1[0][0][0];
    for (int o = tid * 4; o < DD * KC * 16; o += NTHR * 4) {
      *(float4*)(base0 + o) = z;
      *(float4*)(base1 + o) = z;
    }
  }

  v8f acc0 = {};   // D tile, strip cols 0..15
  v8f acc1 = {};   // D tile, strip cols 16..31

  for (int kc = 0; kc < C_DIM; kc += KC) {
    __syncthreads();   // previous chunk fully consumed (and clear visible)

    // ---- stage x2 chunk: strip[di][k][n] = x2[b][kc+k][i+di-4][j0-4+n]
    // Thread (wave=di, lane=k) handles one 32-float row; j0-4 is a multiple
    // of 4 and W%4==0, so every 16B group is aligned and fully in/out.
    {
      const int k = lane;
      const float* src = x2b + ((size_t)(kc + k) * H_DIM + i2) * W_DIM;
#pragma unroll
      for (int g = 0; g < 8; ++g) {
        const int n  = g * 4;
        const int jj = j0 - RAD + n;
        const bool ok = rowValid && ((unsigned)jj < (unsigned)W_DIM);
        float* dst = (n < 16) ? &sB0[wave][k][n] : &sB1[wave][k][n - 16];
#ifdef HAVE_ASYNC_LDS
        if (ok) async_cp16(src + jj, dst);
#else
        float4 v = ok ? *(const float4*)(src + jj)
                      : make_float4(0.f, 0.f, 0.f, 0.f);
        dst[0] = v.x; dst[1] = v.y; dst[2] = v.z; dst[3] = v.w;
#endif
      }
    }

    // ---- stage x1 chunk (transpose): sA[m][k] = x1[b][kc+k][i][j0+m]
    if (tid < 256) {
      const int k = tid >> 4;
      const int m = tid & 15;
      sA[m][k] = x1b[((size_t)(kc + k) * H_DIM + i) * W_DIM + (j0 + m)];
    }

#ifdef HAVE_ASYNC_LDS
    wait_async0();     // own async LDS writes complete before the barrier
#endif
    __syncthreads();

    // ---- compute: wave w handles di = w; two 16x16 N-tiles, K-chunk of 32.
    // A lanes 0-15: (k,k+1) of row m; lanes 16-31: (k+2,k+3).  B mirrors with
    // column n = lane%16 (+16 for tile 1).  Each v2f's elements sit at
    // offsets {0, +16 floats} inside one half-array; the two half-arrays are
    // ~18KB apart, beyond the 8-bit 2addr offset field, so the merger must
    // emit ds_load_2addr_b32 v[a:a+1] ... offset1:16 — the pair lands
    // directly in the consecutive VGPRs WMMA consumes.
    if (rowValid) {
      const float* p0 = &sB0[wave][0][0];
      const float* p1 = &sB1[wave][0][0];
#pragma unroll
      for (int kk = 0; kk < KC; kk += 4) {
        const int kb   = kk + 2 * hi;
        const int base = kb * 16 + mm;
        v2f a;
        a.x = sA[mm][kb + 0];
        a.y = sA[mm][kb + 1];
        v2f b0, b1;
        b0.x = p0[base];
        b0.y = p0[base + 16];
        b1.x = p1[base];
        b1.y = p1[base + 16];
        acc0 = __builtin_amdgcn_wmma_f32_16x16x4_f32(
            false, a, false, b0, (short)0, acc0, false, false);
        acc1 = __builtin_amdgcn_wmma_f32_16x16x4_f32(
            false, a, false, b1, (short)0, acc1, false, false);
      }
    }
  }

  // ---- spill D tiles to LDS for the banded gather.
  // D layout: vgpr v, lanes 0-15 -> M=v, N=lane; lanes 16-31 -> M=8+v.
#pragma unroll
  for (int v = 0; v < 8; ++v) {
    sD[wave][v + 8 * hi][mm]      = acc0[v];
    sD[wave][v + 8 * hi][16 + mm] = acc1[v];
  }
  __syncthreads();

  // ---- gather band n = m + dj and write out (coalesced over m).
  const float scale = 1.0f / (float)C_DIM;
  for (int o = tid; o < DD * DD * TM; o += NTHR) {
    const int m  = o & 15;
    const int d  = o >> 4;       // 0..80 = di*9 + dj
    const int di = d / DD;
    const int dj = d - di * DD;
    const float val = sD[di][m][m + dj] * scale;
    out[(((size_t)b * (DD * DD) + d) * H_DIM + i) * W_DIM + (j0 + m)] = val;
  }
}

extern "C" void kernel_launch(void* const* d_in, const int* in_sizes, int n_in,
                              void* d_out, int out_size, void* d_ws, size_t ws_size,
                              hipStream_t stream) {
  (void)in_sizes; (void)n_in; (void)d_ws; (void)ws_size; (void)out_size;
  const float* x1 = (const float*)d_in[0];
  const float* x2 = (const float*)d_in[1];
  float* out = (float*)d_out;
  dim3 grid(W_DIM / TM, H_DIM, B_DIM);   // (8, 96, 8)
  dim3 block(NTHR);                       // 9 waves: one per di
  corr_wmma_kernel<<<grid, block, 0, stream>>>(x1, x2, out);
}